// BAFM_BRB_65249143161427
// MI455X (gfx1250) — compile-verified
//
#include <hip/hip_runtime.h>
#include <stdint.h>

typedef __attribute__((ext_vector_type(2))) float v2f;
typedef __attribute__((ext_vector_type(8))) float v8f;

#define DSZ  256    // downsampled edge (2048/8)
#define NIMG 8
#define FULL 2048
#define SS   8

// ---------------------------------------------------------------------------
// Inclusive cumsum of 256 floats (viewed as a 16x16 row-major matrix X) using
// chained V_WMMA_F32_16X16X4_F32:
//   inc = Lstrict * R + X * U
// where U[k][j] = (k <= j)  -> intra-row prefix sums,
//       Lstrict[i][k] = (k < i), R[k][*] = rowsum_k -> carry of prior rows.
// Both products share one f32 accumulator across 8 chained K=4 WMMA ops.
// Executed by ONE full wave (lanes 0..31, EXEC all ones).
// A-frag (16x4 f32): lane L -> M = L&15;  VGPRs hold K = {0,1} (L<16) / {2,3}.
// C/D-frag (16x16 f32): VGPR v, lane L -> M = v + 8*(L>>4), N = L&15.
// ---------------------------------------------------------------------------
__device__ __forceinline__ void wmma_cumsum256(const float* sx, float* si,
                                               float* st, int lane) {
  const int m    = lane & 15;
  const int half = lane >> 4;
  const int n    = m;          // B/C/D column index for this lane
  v8f acc = {};
  // P = X * U   (intra-row inclusive prefix)
#pragma unroll
  for (int q = 0; q < 4; ++q) {
    const int k0 = 4 * q + 2 * half;
    v2f a, b;
    a.x = sx[m * 16 + k0];
    a.y = sx[m * 16 + k0 + 1];
    b.x = (k0     <= n) ? 1.0f : 0.0f;
    b.y = (k0 + 1 <= n) ? 1.0f : 0.0f;
    acc = __builtin_amdgcn_wmma_f32_16x16x4_f32(false, a, false, b,
                                                (short)0, acc, false, false);
  }
  // spill P so every lane can read rowsum_r = P[r][15]
#pragma unroll
  for (int v = 0; v < 8; ++v) st[(v + 8 * half) * 16 + n] = acc[v];
  // acc += Lstrict * R   (add sums of all previous rows)
#pragma unroll
  for (int q = 0; q < 4; ++q) {
    const int k0 = 4 * q + 2 * half;
    v2f a, b;
    a.x = (k0     < m) ? 1.0f : 0.0f;
    a.y = (k0 + 1 < m) ? 1.0f : 0.0f;
    b.x = st[k0 * 16 + 15];
    b.y = st[(k0 + 1) * 16 + 15];
    acc = __builtin_amdgcn_wmma_f32_16x16x4_f32(false, a, false, b,
                                                (short)0, acc, false, false);
  }
#pragma unroll
  for (int v = 0; v < 8; ++v) si[(v + 8 * half) * 16 + n] = acc[v];
}

// ---------------------------------------------------------------------------
// One directional segmented-mean pair (forward + backward) over a 256-element
// line held in LDS. Returns lr[t] + rl[t]. 256 threads/block.
// prev/nxt boundary indices via LDS Hillis-Steele cummax / reverse cummin;
// prefix sums via wave-0 WMMA.
// ---------------------------------------------------------------------------
__device__ __forceinline__ float seg_dir_sum(float* sx, int* sp, int* sn,
                                             float* si, float* st,
                                             int zt, int t) {
  sp[t] = zt ? t : -1;     // cummax source
  sn[t] = zt ? t : DSZ;    // reverse-cummin source
  __syncthreads();
#pragma unroll
  for (int off = 1; off < DSZ; off <<= 1) {
    const int vmax = sp[t];
    const int umax = (t >= off) ? sp[t - off] : -1;
    const int vmin = sn[t];
    const int umin = (t + off < DSZ) ? sn[t + off] : DSZ;
    __syncthreads();
    sp[t] = vmax > umax ? vmax : umax;
    sn[t] = vmin < umin ? vmin : umin;
    __syncthreads();
  }
  if (t < 32) wmma_cumsum256(sx, si, st, t);
  __syncthreads();

  const int prev = (t > 0)       ? sp[t - 1] : -1;  // last boundary before t
  const int nxt  = (t < DSZ - 1) ? sn[t + 1] : DSZ; // first boundary after t
  const float x  = sx[t];
  float out;
  if (!zt && prev >= 0 && nxt < DSZ) {
    const float inc_t   = si[t];
    const float inc_pm1 = (prev > 0) ? si[prev - 1] : 0.0f;
    const float inc_tm1 = (t > 0)    ? si[t - 1]    : 0.0f;
    const float inc_n   = si[nxt];
    out = (inc_t - inc_pm1) / (float)(t - prev + 1)     // lr
        + (inc_n - inc_tm1) / (float)(nxt - t + 1);     // rl
  } else {
    out = x + x;  // both directions fall back to x[t]
  }
  return out;
}

// ---------------------------------------------------------------------------
// Kernel 1: downsample x0 / (x1==0) into compact planes (L2-resident, 2MB ea)
// ---------------------------------------------------------------------------
__global__ void __launch_bounds__(256)
gather_kernel(const float* __restrict__ x0, const int* __restrict__ x1,
              float* __restrict__ x0d, int* __restrict__ zd) {
  const int i = blockIdx.x * 256 + threadIdx.x;  // 0 .. 8*256*256-1
  const int n = i >> 16;
  const int r = (i >> 8) & 255;
  const int c = i & 255;
  const long long src =
      ((long long)n * FULL + (long long)r * SS) * FULL + (long long)c * SS;
  x0d[i] = x0[src];
  zd[i]  = (x1[src] == 0) ? 1 : 0;
}

// ---------------------------------------------------------------------------
// Kernel 2: row-direction segmented means. One block per (n, row).
// Row is staged into LDS with the gfx1250 async global->LDS path.
// ---------------------------------------------------------------------------
__global__ void __launch_bounds__(256)
row_scan_kernel(const float* __restrict__ x0d, const int* __restrict__ zd,
                float* __restrict__ rowpart) {
  __shared__ float sx[DSZ];
  __shared__ float si[DSZ];
  __shared__ float st[DSZ];
  __shared__ int   sp[DSZ];
  __shared__ int   sn[DSZ];
  const int t    = threadIdx.x;
  const int base = blockIdx.x * DSZ;           // blockIdx = n*256 + r

  // async copy of this wave's row elements directly into LDS
  // (flat shared pointer low 32 bits == LDS byte address)
  const unsigned lx =
      (unsigned)(uintptr_t)(void*)(&sx[t]);
  const unsigned long long gx =
      (unsigned long long)(uintptr_t)(x0d + base + t);
  asm volatile("global_load_async_to_lds_b32 %0, %1, off"
               :: "v"(lx), "v"(gx) : "memory");
  const int zt = zd[base + t];                 // overlap with async copy
  asm volatile("s_wait_asynccnt 0x0" ::: "memory");
  __syncthreads();

  const float out = seg_dir_sum(sx, sp, sn, si, st, zt, t);
  rowpart[base + t] = out;
}

// ---------------------------------------------------------------------------
// Kernel 3: column-direction segmented means + combine. One block per (n,col).
// ---------------------------------------------------------------------------
__global__ void __launch_bounds__(256)
col_scan_kernel(const float* __restrict__ x0d, const int* __restrict__ zd,
                const float* __restrict__ rowpart, float* __restrict__ yd) {
  __shared__ float sx[DSZ];
  __shared__ float si[DSZ];
  __shared__ float st[DSZ];
  __shared__ int   sp[DSZ];
  __shared__ int   sn[DSZ];
  const int t   = threadIdx.x;
  const int n   = blockIdx.x >> 8;
  const int c   = blockIdx.x & 255;
  const int idx = (n << 16) + t * DSZ + c;     // column walk (L2-resident)
  sx[t] = x0d[idx];
  const int zt = zd[idx];
  // seg_dir_sum's first barrier publishes sx to the whole block
  const float out = seg_dir_sum(sx, sp, sn, si, st, zt, t);
  yd[idx] = rowpart[idx] + out;                // lr+rl+tb+bt at downsampled res
}

// ---------------------------------------------------------------------------
// Kernel 4: 8x8 nearest upsample. The bandwidth-critical 134MB store:
// one aligned float4 per thread, fully coalesced (4 | 8 so each float4 is a
// single repeated value).
// ---------------------------------------------------------------------------
__global__ void __launch_bounds__(256)
upsample_kernel(const float* __restrict__ yd, float4* __restrict__ out) {
  const long long i = (long long)blockIdx.x * 256 + threadIdx.x; // float4 idx
  const int c4 = (int)(i & 511);        // 2048/4 float4 per row
  const long long rest = i >> 9;
  const int R = (int)(rest & 2047);
  const int n = (int)(rest >> 11);
  const float v = yd[(n << 16) + ((R >> 3) << 8) + (c4 >> 1)];
  out[i] = make_float4(v, v, v, v);
}

// ---------------------------------------------------------------------------
extern "C" void kernel_launch(void* const* d_in, const int* in_sizes, int n_in,
                              void* d_out, int out_size, void* d_ws,
                              size_t ws_size, hipStream_t stream) {
  const float* x0 = (const float*)d_in[0];
  const int*   x1 = (const int*)d_in[1];

  const int nds = NIMG * DSZ * DSZ;            // 524288 downsampled elems
  float* x0d     = (float*)d_ws;               // 2 MB
  int*   zd      = (int*)(x0d + nds);          // 2 MB
  float* rowpart = (float*)(zd + nds);         // 2 MB
  float* yd      = rowpart + nds;              // 2 MB   (8 MB total in ws)

  gather_kernel  <<<nds / 256,  256, 0, stream>>>(x0, x1, x0d, zd);
  row_scan_kernel<<<NIMG * DSZ, 256, 0, stream>>>(x0d, zd, rowpart);
  col_scan_kernel<<<NIMG * DSZ, 256, 0, stream>>>(x0d, zd, rowpart, yd);

  const long long total4 = (long long)NIMG * FULL * FULL / 4;  // 8.39M
  upsample_kernel<<<(int)(total4 / 256), 256, 0, stream>>>(yd, (float4*)d_out);
}